// MMILayerMatrix_Even_35364760715680
// MI455X (gfx1250) — compile-verified
//
#include <hip/hip_runtime.h>

// MMI block-diagonal complex 2x2 layer as f32 WMMA on gfx1250 (CDNA5).
// Y[16,cols] = M16[16,16] @ X16[16,cols], M16 = blockdiag of four 4x4 real
// blocks Q_m acting on interleaved (a0,b0,a1,b1) rows:
//   Q = [[ t, 0, 0,-k],
//        [ 0, t, k, 0],
//        [ 0,-k, t, 0],
//        [ k, 0, 0, t]]
// Done with 4 chained V_WMMA_F32_16X16X4_F32 per 16x16 column tile.
// Bandwidth-bound: ~1.07 GB traffic -> ~46 us floor at 23.3 TB/s.

typedef __attribute__((ext_vector_type(2))) float v2f;
typedef __attribute__((ext_vector_type(8))) float v8f;

#define MMI_COLS 16384
#define MMI_NROWS 4096
#define MMI_NMMI 2048
#define MMI_TPW 4            // column tiles per wave
#define MMI_WAVES 8          // waves per block (256 threads, wave32)

__device__ __forceinline__ float mmi_qval(int i, int j, float t, float k) {
    // 4x4 real MMI block entry Q[i][j]
    if (i == j) return t;
    if (i == 0 && j == 3) return -k;
    if (i == 1 && j == 2) return  k;
    if (i == 2 && j == 1) return -k;
    if (i == 3 && j == 0) return  k;
    return 0.0f;
}

__global__ __launch_bounds__(256)
void MMILayerMatrix_Even_wmma_kernel(const float* __restrict__ xr,
                                     const float* __restrict__ xi,
                                     const float* __restrict__ loss,
                                     const float* __restrict__ imb,
                                     float* __restrict__ out) {
    const int lane  = threadIdx.x & 31;
    const int wave  = threadIdx.x >> 5;
    const int laneN = lane & 15;    // column within tile / B & D lane index
    const int laneH = lane >> 4;    // half-wave selector

    const int g  = blockIdx.y;      // group of 4 MMIs = 16 interleaved rows
    const int m0 = g * 4;           // first MMI of group
    const int r0 = g * 8;           // first physical input/output row

    // --- per-group coefficients (uniform across block -> scalar loads) ---
    float tt[4], kk[4];
#pragma unroll
    for (int s = 0; s < 4; ++s) {
        float gi    = imb[m0 + s];
        float alpha = 0.5f * (gi - 1.0f) / (gi + 1.0f);
        float sq    = sqrtf(loss[m0 + s]);
        tt[s] = sq * sqrtf(0.5f + alpha);   // tau_tf (real)
        kk[s] = sq * sqrtf(0.5f - alpha);   // |kappa_tf| (kappa = i*k)
    }

    // --- A operands: K-slices of block-diag M16, built once, reused ---
    // A 16x4 f32 layout: lane = M row (mod 16); VGPR v holds K = v + 2*laneH.
    // Slice s (K = 4s..4s+3) is nonzero only for rows 4s..4s+3 (block diag).
    v2f A[4];
#pragma unroll
    for (int s = 0; s < 4; ++s) {
        int   i  = laneN - 4 * s;       // row within 4x4 block
        int   j0 = 2 * laneH;           // K-within-slice for VGPR0
        float ax = 0.0f, ay = 0.0f;
        if (i >= 0 && i < 4) {
            ax = mmi_qval(i, j0,     tt[s], kk[s]);
            ay = mmi_qval(i, j0 + 1, tt[s], kk[s]);
        }
        A[s].x = ax;
        A[s].y = ay;
    }

    // --- column tiles ---
    const int tileBase = (blockIdx.x * MMI_WAVES + wave) * MMI_TPW;
#pragma unroll
    for (int t = 0; t < MMI_TPW; ++t) {
        const int col = (tileBase + t) * 16 + laneN;

        // B 4x16 f32 layout (mirrors A): N = laneN, K = vgpr + 2*laneH.
        // Slice s rows: K0=a_{2m}, K1=b_{2m}, K2=a_{2m+1}, K3=b_{2m+1}
        //  -> B.x (VGPR0) = x_real[2m+laneH], B.y (VGPR1) = x_imag[2m+laneH]
        v2f B[4];
#pragma unroll
        for (int s = 0; s < 4; ++s) {
            const long long off =
                (long long)(2 * (m0 + s) + laneH) * MMI_COLS + col;
            B[s].x = xr[off];
            B[s].y = xi[off];
        }

        v8f c = {0.f, 0.f, 0.f, 0.f, 0.f, 0.f, 0.f, 0.f};
#pragma unroll
        for (int s = 0; s < 4; ++s) {
            // 8 args: (neg_a, A, neg_b, B, c_mod, C, reuse_a, reuse_b)
            c = __builtin_amdgcn_wmma_f32_16x16x4_f32(
                    false, A[s], false, B[s], (short)0, c, false, false);
        }

        // D layout: VGPR v = Y16 row v (lanes 0-15) / v+8 (lanes 16-31).
        // Y16 row pairs (2p,2p+1) = (real,imag) of output row rowBase+p
        // -> coalesced 64-bit interleaved-complex stores.
        const int rowBase = r0 + 4 * laneH;
#pragma unroll
        for (int p = 0; p < 4; ++p) {
            const long long o =
                ((long long)(rowBase + p) * MMI_COLS + col) * 2;
            float2 v;
            v.x = c[2 * p];
            v.y = c[2 * p + 1];
            *reinterpret_cast<float2*>(out + o) = v;
        }
    }
}

extern "C" void kernel_launch(void* const* d_in, const int* in_sizes, int n_in,
                              void* d_out, int out_size, void* d_ws, size_t ws_size,
                              hipStream_t stream) {
    (void)in_sizes; (void)n_in; (void)out_size; (void)d_ws; (void)ws_size;
    const float* xr   = (const float*)d_in[0];   // x_real   [4096,16384] f32
    const float* xi   = (const float*)d_in[1];   // x_imag   [4096,16384] f32
    const float* loss = (const float*)d_in[2];   // mmi_i_losses   [2048] f32
    const float* imb  = (const float*)d_in[3];   // mmi_imbalances [2048] f32
    float* out = (float*)d_out;                  // complex64 interleaved [4096,16384]

    // 1024 col tiles / (8 waves * 4 tiles) = 32 blocks in x; 512 row groups in y
    dim3 grid(32, 512);
    MMILayerMatrix_Even_wmma_kernel<<<grid, 256, 0, stream>>>(xr, xi, loss, imb, out);
}